// AddAttention_53386443489403
// MI455X (gfx1250) — compile-verified
//
#include <hip/hip_runtime.h>
#include <hip/hip_bf16.h>

typedef float v2f __attribute__((ext_vector_type(2)));
typedef float v4f __attribute__((ext_vector_type(4)));
typedef float v8f __attribute__((ext_vector_type(8)));

#define NEG_MASK (-1e6f)

__device__ __forceinline__ float fast_tanh(float x) {
#if __has_builtin(__builtin_amdgcn_tanhf)
  return __builtin_amdgcn_tanhf(x);
#else
  float y;
  // native gfx1250 transcendental; v_nop covers the TRANS result-use hazard
  asm volatile("v_tanh_f32 %0, %1\n\tv_nop" : "=v"(y) : "v"(x));
  return y;
#endif
}

__device__ __forceinline__ float wave_max(float v) {
  for (int o = 16; o > 0; o >>= 1) v = fmaxf(v, __shfl_xor(v, o, 32));
  return v;
}
__device__ __forceinline__ float wave_sum(float v) {
  for (int o = 16; o > 0; o >>= 1) v += __shfl_xor(v, o, 32);
  return v;
}

// Async copy of 16 bytes global -> LDS (gfx1250, tracked by ASYNCcnt).
__device__ __forceinline__ void async_g2lds_b128(void* lds_ptr, const void* gptr) {
  const unsigned lds_off = (unsigned)(unsigned long long)lds_ptr;  // LDS aperture: offset in addr[31:0]
  const unsigned long long ga = (unsigned long long)gptr;
  asm volatile("global_load_async_to_lds_b128 %0, %1, off"
               :: "v"(lds_off), "v"(ga) : "memory");
}
__device__ __forceinline__ void wait_async0() {
  asm volatile("s_wait_asynccnt 0x0" ::: "memory");
}

// ---------------------------------------------------------------------------
// Projection: OUT[b, 16 rows, H=128] = X[b, 16 rows, 256] @ W[256, 128]
// One block per (b, row-tile). 8 waves; wave w owns output columns w*16..w*16+15.
// f32 WMMA 16x16x4, K-loop of 64 steps.
// ---------------------------------------------------------------------------
__global__ __launch_bounds__(256)
void proj16x128(const float* __restrict__ X, const float* __restrict__ W,
                float* __restrict__ OUT) {
  constexpr int D = 256, H = 128;
  constexpr int XS = 260;                 // stride (floats), %4==0, conflict-free A reads
  __shared__ __align__(16) float xt[16 * XS];

  const int b   = blockIdx.x >> 4;
  const int rt  = blockIdx.x & 15;
  const int tid = threadIdx.x;

  const float* Xb = X + (size_t)(b * 256 + rt * 16) * D;
  for (int i = tid; i < (16 * D) / 4; i += 256) {
    const int elem = i * 4;
    *(v4f*)&xt[(elem >> 8) * XS + (elem & 255)] = *(const v4f*)&Xb[elem];
  }
  __syncthreads();

  const int wave = tid >> 5, lane = tid & 31;
  const int row  = lane & 15;
  const int khi  = (lane >> 4) * 2;       // lanes 16-31 hold K+2, K+3
  const int col  = wave * 16 + row;       // B/D column (N)

  v8f acc = {0.f, 0.f, 0.f, 0.f, 0.f, 0.f, 0.f, 0.f};
  for (int k = 0; k < D; k += 4) {
    const int kb = k + khi;
    v2f a, bf;
    a.x  = xt[row * XS + kb];
    a.y  = xt[row * XS + kb + 1];
    bf.x = W[(size_t)kb * H + col];
    bf.y = W[(size_t)(kb + 1) * H + col];
    acc = __builtin_amdgcn_wmma_f32_16x16x4_f32(false, a, false, bf,
                                                (short)0, acc, false, false);
  }

  float* Ob = OUT + (size_t)(b * 256 + rt * 16) * H;
  const int mhi = (lane >> 4) * 8;        // lanes 16-31 hold rows M=8..15
  for (int i = 0; i < 8; ++i)
    Ob[(size_t)(i + mhi) * H + col] = acc[i];
}

// ---------------------------------------------------------------------------
// Fused scores -> masked softmax -> attn @ V, one block per (b, 16-q tile).
// ---------------------------------------------------------------------------
__global__ __launch_bounds__(256)
void attn_kernel(const float* __restrict__ Qh, const float* __restrict__ Kh,
                 const float* __restrict__ V,  const float* __restrict__ Wv,
                 const int* __restrict__ vlen, float* __restrict__ OUT) {
  constexpr int H = 128, KK = 256, DV = 256;
  constexpr int QS = 132, KS = 132, SS = 260;   // all %4==0 -> 16B-aligned rows
  __shared__ __align__(16) float qh[16 * QS];
  __shared__ __align__(16) float kh[32 * KS];
  __shared__ __align__(16) float wv[H];
  __shared__ __align__(16) float sc[16 * SS];

  const int b   = blockIdx.x >> 4;
  const int qt  = blockIdx.x & 15;
  const int tid = threadIdx.x;
  const int wave = tid >> 5, lane = tid & 31;

  const float* Qb = Qh + (size_t)(b * 256 + qt * 16) * H;
  for (int i = tid; i < (16 * H) / 4; i += 256) {
    const int elem = i * 4;
    *(v4f*)&qh[(elem >> 7) * QS + (elem & 127)] = *(const v4f*)&Qb[elem];
  }
  if (tid < H) wv[tid] = Wv[tid];
  __syncthreads();

  // ---- scores: wave w, lane l -> rows {w, w+8}, key kc+l (kh read shared) ----
  const float* Kb = Kh + (size_t)b * 256 * H;
  for (int kc = 0; kc < KK; kc += 32) {
    // async-stage kh[32][128] (16 KB) straight into LDS, b128 per op
    for (int i = tid; i < (32 * H) / 4; i += 256) {
      const int elem = i * 4;
      async_g2lds_b128(&kh[(elem >> 7) * KS + (elem & 127)],
                       Kb + (size_t)kc * H + elem);
    }
    wait_async0();
    __syncthreads();

    float s0 = 0.f, s1 = 0.f;
    const float* q0r = &qh[wave * QS];
    const float* q1r = &qh[(wave + 8) * QS];
    const float* kr  = &kh[lane * KS];
    for (int h = 0; h < H; h += 4) {
      const v4f kv = *(const v4f*)&kr[h];
      const v4f wq = *(const v4f*)&wv[h];
      const v4f qa = *(const v4f*)&q0r[h];
      const v4f qb = *(const v4f*)&q1r[h];
      s0 += wq.x * fast_tanh(qa.x + kv.x);
      s1 += wq.x * fast_tanh(qb.x + kv.x);
      s0 += wq.y * fast_tanh(qa.y + kv.y);
      s1 += wq.y * fast_tanh(qb.y + kv.y);
      s0 += wq.z * fast_tanh(qa.z + kv.z);
      s1 += wq.z * fast_tanh(qb.z + kv.z);
      s0 += wq.w * fast_tanh(qa.w + kv.w);
      s1 += wq.w * fast_tanh(qb.w + kv.w);
    }
    sc[wave * SS + kc + lane]       = s0;
    sc[(wave + 8) * SS + kc + lane] = s1;
    __syncthreads();
  }

  // ---- masked softmax over K (each wave normalizes the rows it produced) ----
  const int valid = vlen[b];
  for (int rr = 0; rr < 2; ++rr) {
    const int row = wave + rr * 8;
    float x[8];
    float m = -3.0e38f;
    for (int i = 0; i < 8; ++i) {
      const int kidx = lane + 32 * i;
      float v = sc[row * SS + kidx];
      if (kidx >= valid) v = NEG_MASK;        // matches reference where(mask, s, -1e6)
      x[i] = v;
      m = fmaxf(m, v);
    }
    m = wave_max(m);
    float s = 0.f;
    for (int i = 0; i < 8; ++i) { x[i] = __expf(x[i] - m); s += x[i]; }
    s = wave_sum(s);
    const float inv = 1.0f / s;
    for (int i = 0; i < 8; ++i)
      sc[row * SS + lane + 32 * i] = x[i] * inv;
  }
  __syncthreads();

  // ---- O[16 x 256] = attn[16 x 256] @ V[256 x 256] via f32 WMMA ----
  const int row16 = lane & 15;
  const int khi   = (lane >> 4) * 2;
  const float* Vb = V + (size_t)b * 256 * DV;
  float* Ob = OUT + (size_t)(b * 256 + qt * 16) * DV;
  for (int t = 0; t < 2; ++t) {
    const int col = (wave * 2 + t) * 16 + row16;
    v8f acc = {0.f, 0.f, 0.f, 0.f, 0.f, 0.f, 0.f, 0.f};
    for (int k = 0; k < KK; k += 4) {
      const int kb = k + khi;
      v2f a, bf;
      a.x  = sc[row16 * SS + kb];
      a.y  = sc[row16 * SS + kb + 1];
      bf.x = Vb[(size_t)kb * DV + col];
      bf.y = Vb[(size_t)(kb + 1) * DV + col];
      acc = __builtin_amdgcn_wmma_f32_16x16x4_f32(false, a, false, bf,
                                                  (short)0, acc, false, false);
    }
    const int mhi = (lane >> 4) * 8;
    for (int i = 0; i < 8; ++i)
      Ob[(size_t)(i + mhi) * DV + col] = acc[i];
  }
}

// ---------------------------------------------------------------------------
extern "C" void kernel_launch(void* const* d_in, const int* in_sizes, int n_in,
                              void* d_out, int out_size, void* d_ws, size_t ws_size,
                              hipStream_t stream) {
  const float* query = (const float*)d_in[0];  // [16,256,256]
  const float* key   = (const float*)d_in[1];  // [16,256,256]
  const float* value = (const float*)d_in[2];  // [16,256,256]
  const float* Wq    = (const float*)d_in[3];  // [256,128]
  const float* Wk    = (const float*)d_in[4];  // [256,128]
  const float* Wv    = (const float*)d_in[5];  // [128]
  const int*   vl    = (const int*)d_in[6];    // [16] (JAX x64 off -> int32)
  float* out = (float*)d_out;                  // [16,256,256]

  float* qh_ws = (float*)d_ws;                           // [16,256,128] f32 = 2 MB
  float* kh_ws = qh_ws + (size_t)16 * 256 * 128;         // [16,256,128] f32 = 2 MB

  const dim3 blk(256);
  proj16x128<<<dim3(16 * 16), blk, 0, stream>>>(query, Wq, qh_ws);
  proj16x128<<<dim3(16 * 16), blk, 0, stream>>>(key,   Wk, kh_ws);
  attn_kernel<<<dim3(16 * 16), blk, 0, stream>>>(qh_ws, kh_ws, value, Wv, vl, out);
}